// ConsecutiveLoss_69337952027144
// MI455X (gfx1250) — compile-verified
//
#include <hip/hip_runtime.h>
#include <stdint.h>

// Problem shape (from reference): x is [bsz, 8192] fp32; bsz = total/8192.
#define SEQ     8192
#define THREADS 256
#define PER_T   (SEQ / THREADS)           // 32 floats per thread
#define ASYNC_ITERS (SEQ / (THREADS * 4)) // 8 x b128 issues per thread

typedef float v2f __attribute__((ext_vector_type(2)));
typedef float v8f __attribute__((ext_vector_type(8)));

// ---------------------------------------------------------------------------
// Kernel 1: one workgroup per row.
//   - async-DMA the whole 32KB row into LDS (global_load_async_to_lds_b128,
//     ASYNCcnt-tracked, drained with s_wait_asynccnt)
//   - pass 1 (from LDS): nonzero count L  (integer LDS atomic: deterministic)
//   - pass 2 (from LDS): sum |x[p]-x[p-1]| for p in [1, L-1]
//   - fixed-order LDS tree reduction (deterministic), write per-row value
// Single HBM pass per element -> bandwidth-optimal (~5.8us floor @ 23.3TB/s).
// ---------------------------------------------------------------------------
__global__ __launch_bounds__(THREADS)
void consecutive_loss_row_kernel(const float* __restrict__ x,
                                 float* __restrict__ row_out)
{
    __shared__ float srow[SEQ];       // 32 KB row staging
    __shared__ float fpart[THREADS];  // float reduction scratch
    __shared__ int   cnt;

    const int row = blockIdx.x;
    const int tid = threadIdx.x;

    if (tid == 0) cnt = 0;

    // ---- async copy: global -> LDS, 16 bytes per lane per issue ----
    const float* gbase = x + (size_t)row * SEQ;
    #pragma unroll
    for (int i = 0; i < ASYNC_ITERS; ++i) {
        const int fidx = (i * THREADS + tid) * 4;           // float index
        const float* g = gbase + fidx;
        // low 32 bits of a generic LDS pointer == LDS byte offset
        unsigned lds_off = (unsigned)(uintptr_t)(&srow[fidx]);
        asm volatile("global_load_async_to_lds_b128 %0, %1, off"
                     :: "v"(lds_off), "v"(g)
                     : "memory");
    }
    asm volatile("s_wait_asynccnt 0" ::: "memory");
    __syncthreads();   // all waves' async data visible in LDS

    // ---- pass 1: nonzero count (order-independent integer atomic) ----
    const int base = tid * PER_T;
    int c = 0;
    #pragma unroll
    for (int j = 0; j < PER_T; ++j)
        c += (srow[base + j] != 0.0f) ? 1 : 0;
    atomicAdd(&cnt, c);
    __syncthreads();
    const int L = cnt;

    // ---- pass 2: masked |diff| sum, positions p in [1, L-1] ----
    float s = 0.0f;
    #pragma unroll
    for (int j = 0; j < PER_T; ++j) {
        const int p = base + j;
        float d = (p >= 1 && p < L) ? fabsf(srow[p] - srow[p - 1]) : 0.0f;
        s += d;
    }
    fpart[tid] = s;
    __syncthreads();

    // fixed-order tree reduction -> deterministic
    #pragma unroll
    for (int st = THREADS / 2; st > 0; st >>= 1) {
        if (tid < st) fpart[tid] += fpart[tid + st];
        __syncthreads();
    }

    if (tid == 0)
        row_out[row] = (row >= 1 && L > 0) ? (fpart[0] / (float)L) : 0.0f;
}

// ---------------------------------------------------------------------------
// Kernel 2: final reduction of per-row values using V_WMMA_F32_16X16X4_F32.
// A = ones(16x4)  =>  D[m,n] = sum_k B[k,n] + C[m,n].
// Feed 64 values per WMMA through B and accumulate in C; the sum of ALL 256
// entries of the final C equals 16 * sum(values) regardless of the exact
// B-operand lane layout (permutation-invariant).
//
// Hot loop handles full 64-element chunks with UNGUARDED float2 (b64) loads
// (EXEC stays all-ones -> WMMA-legal, no branchy cndmask/exec save-restore);
// a guarded tail covers n % 64 != 0 (empty for the reference shape).
// ---------------------------------------------------------------------------
__global__ __launch_bounds__(32)
void consecutive_loss_reduce_kernel(const float* __restrict__ rowv,
                                    float* __restrict__ out,
                                    int n, float inv_bsz)
{
    const int lane = threadIdx.x;

    v8f c = {};
    v2f a;
    a.x = 1.0f;
    a.y = 1.0f;

    const int full = n / 64;                 // full 64-value chunks
    const float2* rowv2 = (const float2*)rowv;

    for (int it = 0; it < full; ++it) {
        float2 p = rowv2[it * 32 + lane];    // global_load_b64, no guards
        v2f b;
        b.x = p.x;
        b.y = p.y;
        // 8 args: (neg_a, A, neg_b, B, c_mod, C, reuse_a, reuse_b)
        c = __builtin_amdgcn_wmma_f32_16x16x4_f32(
                false, a, false, b, (short)0, c, false, false);
    }

    // tail (n % 64 values) -- guarded; empty for bsz = 4096
    const int tail_base = full * 64;
    if (tail_base < n) {
        const int i0 = tail_base + lane * 2;
        v2f b;
        b.x = (i0     < n) ? rowv[i0]     : 0.0f;
        b.y = (i0 + 1 < n) ? rowv[i0 + 1] : 0.0f;
        c = __builtin_amdgcn_wmma_f32_16x16x4_f32(
                false, a, false, b, (short)0, c, false, false);
    }

    // per-lane sum of the 8 accumulator VGPRs, then wave32 reduce:
    // total over all 256 D entries == 16 * sum(rowv)
    float t = c[0] + c[1] + c[2] + c[3] + c[4] + c[5] + c[6] + c[7];
    #pragma unroll
    for (int off = 16; off > 0; off >>= 1)
        t += __shfl_xor(t, off, 32);

    if (lane == 0)
        out[0] = t * (1.0f / 16.0f) * inv_bsz;
}

// ---------------------------------------------------------------------------
extern "C" void kernel_launch(void* const* d_in, const int* in_sizes, int n_in,
                              void* d_out, int out_size, void* d_ws, size_t ws_size,
                              hipStream_t stream)
{
    (void)n_in; (void)out_size; (void)ws_size;

    const float* x = (const float*)d_in[0];
    const int total = in_sizes[0];
    const int bsz   = total / SEQ;          // 4096 for the reference shape

    float* rowv = (float*)d_ws;             // bsz floats of scratch

    consecutive_loss_row_kernel<<<bsz, THREADS, 0, stream>>>(x, rowv);
    consecutive_loss_reduce_kernel<<<1, 32, 0, stream>>>(
        rowv, (float*)d_out, bsz, 1.0f / (float)bsz);
}